// MathONet_23562190585920
// MI455X (gfx1250) — compile-verified
//
#include <hip/hip_runtime.h>
#include <hip/hip_bf16.h>

// MathONet forward for MI455X (gfx1250, wave32).
//
// Each hidden layer's per-sample bilinear form is one GEMM:
//   Z[b,j] = sum_kk A[b,kk] * Wf[j,kk],  A[b, i*128+k] = x1[b,i]*x[b,k]
// with Wf = poly_W repacked to [256, HIN*128] row-major, plus a small bias
// GEMM for poly_W's k=128 column. A is generated on the fly from LDS tiles.
// f32 WMMA (16x16x4) keeps full reference precision through chained sin/cos.
//
// Round-2 structure: wave = 2 M-subtiles x 4 N-subtiles (8 accum chains,
// 0.5 vmem-b64 per WMMA, B shared across M), deterministic 4-way K-split
// over the i dimension for occupancy (1024 waves/layer), partials reduced by
// a fused reduce+OperNet kernel. No atomics -> bitwise deterministic.

typedef float v2f __attribute__((ext_vector_type(2)));
typedef float v8f __attribute__((ext_vector_type(8)));

#define E_TOTAL 98304
#define BATCH   2048

__device__ __forceinline__ v8f wmma4(v2f a, v2f b, v8f c) {
  return __builtin_amdgcn_wmma_f32_16x16x4_f32(false, a, false, b, (short)0, c, false, false);
}

// --- repack poly_W [E,129] -> pmain [E,128] (k<128) + pbias [E] (k==128) ---
__global__ __launch_bounds__(256) void repack_kernel(const float* __restrict__ polyW,
                                                     float* __restrict__ pmain,
                                                     float* __restrict__ pbias) {
  const unsigned total = (unsigned)E_TOTAL * 129u;
  for (unsigned t = blockIdx.x * blockDim.x + threadIdx.x; t < total;
       t += gridDim.x * blockDim.x) {
    unsigned e = t / 129u;
    unsigned k = t - e * 129u;
    float v = polyW[t];
    if (k == 128u) pbias[e] = v;
    else           pmain[(size_t)e * 128u + k] = v;
  }
}

// --- partial-Z for one layer, one K-slice: zpart[s][b][j] ---
template <int HIN>
__global__ __launch_bounds__(256) void polylayer_kernel(
    const float* __restrict__ xin,    // [B, HIN]  layer input (x1)
    const float* __restrict__ x,      // [B, 128]  original input
    const float* __restrict__ pmain,  // repacked weights (base pre-offset), [256, HIN*128]
    const float* __restrict__ pbias,  // bias column (base pre-offset), [256, HIN]
    float* __restrict__ zpart)        // [4, B, 256] partial sums
{
  constexpr int SL = HIN / 4;                 // i-slice length (32 or 64)
  constexpr int S1 = SL + 4;                  // LDS stride: bank step 4 mod 64
  __shared__ __align__(16) float s_xin[64 * S1];
  __shared__ __align__(16) float s_x[64 * 132];
  const int Mblock = blockIdx.x * 64;
  const int i0 = blockIdx.y * SL;

  for (int t = threadIdx.x; t < 64 * SL; t += 256) {
    int b = t / SL, c = t & (SL - 1);
    s_xin[b * S1 + c] = xin[(size_t)(Mblock + b) * HIN + i0 + c];
  }
  for (int t = threadIdx.x; t < 64 * 128; t += 256) {
    int b = t >> 7, c = t & 127;
    s_x[b * 132 + c] = x[(size_t)(Mblock + b) * 128 + c];
  }
  __syncthreads();

  const int wave = threadIdx.x >> 5;
  const int lane = threadIdx.x & 31;
  const int l15  = lane & 15;
  const int hi8  = (lane >> 4) * 8;           // C-matrix row offset, upper half-wave
  const int koff = (lane >> 4) * 2;           // A/B hold K=2,3 in lanes 16-31
  const int mp   = wave & 1;                  // which pair of 16-row M-subtiles
  const int ng   = wave >> 1;                 // 64-neuron group (0..3)

  const int r0 = mp * 32 + l15;
  const int r1 = r0 + 16;
  const float* xr0  = s_x + r0 * 132 + koff;
  const float* xr1  = s_x + r1 * 132 + koff;
  const float* xin0 = s_xin + r0 * S1;
  const float* xin1 = s_xin + r1 * S1;

  v8f acc[2][4];
#pragma unroll
  for (int s = 0; s < 2; ++s)
#pragma unroll
    for (int n = 0; n < 4; ++n) acc[s][n] = v8f{};

  const float* bptr[4];
  const float* biasptr[4];
#pragma unroll
  for (int n = 0; n < 4; ++n) {
    int j = ng * 64 + n * 16 + l15;
    bptr[n]    = pmain + (size_t)j * HIN * 128 + (size_t)i0 * 128 + koff;
    biasptr[n] = pbias + (size_t)j * HIN + i0 + koff;
  }

  // Main GEMM slice: kk = i*128 + k, i in [i0, i0+SL)
  for (int ii = 0; ii < SL; ++ii) {
    float a0s = xin0[ii];                     // one ds_load per i, reused 32x
    float a1s = xin1[ii];
    const float* b0 = bptr[0] + ii * 128;
    const float* b1 = bptr[1] + ii * 128;
    const float* b2 = bptr[2] + ii * 128;
    const float* b3 = bptr[3] + ii * 128;
    if (ii + 1 < SL) {                        // warm WGP$ for next weight row
      __builtin_prefetch(b0 + 128, 0, 3);
      __builtin_prefetch(b1 + 128, 0, 3);
      __builtin_prefetch(b2 + 128, 0, 3);
      __builtin_prefetch(b3 + 128, 0, 3);
    }
#pragma unroll 8
    for (int k0 = 0; k0 < 128; k0 += 4) {
      v2f xv0 = *(const v2f*)(xr0 + k0);      // ds_load_b64
      v2f xv1 = *(const v2f*)(xr1 + k0);
      v2f a0 = {a0s * xv0[0], a0s * xv0[1]};  // outer-product A fragments
      v2f a1 = {a1s * xv1[0], a1s * xv1[1]};
      v2f bf;
      bf = *(const v2f*)(b0 + k0);
      acc[0][0] = wmma4(a0, bf, acc[0][0]);
      acc[1][0] = wmma4(a1, bf, acc[1][0]);
      bf = *(const v2f*)(b1 + k0);
      acc[0][1] = wmma4(a0, bf, acc[0][1]);
      acc[1][1] = wmma4(a1, bf, acc[1][1]);
      bf = *(const v2f*)(b2 + k0);
      acc[0][2] = wmma4(a0, bf, acc[0][2]);
      acc[1][2] = wmma4(a1, bf, acc[1][2]);
      bf = *(const v2f*)(b3 + k0);
      acc[0][3] = wmma4(a0, bf, acc[0][3]);
      acc[1][3] = wmma4(a1, bf, acc[1][3]);
    }
  }

  // Bias column slice (the "+1" of x_aug): Z += x1[:, i0:i0+SL] @ pbias_block^T
#pragma unroll 4
  for (int ii = 0; ii < SL; ii += 4) {
    v2f a0 = *(const v2f*)(xin0 + ii + koff);
    v2f a1 = *(const v2f*)(xin1 + ii + koff);
#pragma unroll
    for (int n = 0; n < 4; ++n) {
      v2f bf = *(const v2f*)(biasptr[n] + ii);
      acc[0][n] = wmma4(a0, bf, acc[0][n]);
      acc[1][n] = wmma4(a1, bf, acc[1][n]);
    }
  }

  // Store partial sums (no activation here; reduce kernel finishes)
  float* zp = zpart + (size_t)blockIdx.y * (BATCH * 256);
#pragma unroll
  for (int s = 0; s < 2; ++s)
#pragma unroll
    for (int n = 0; n < 4; ++n) {
      int j = ng * 64 + n * 16 + l15;
#pragma unroll
      for (int r = 0; r < 8; ++r) {
        int b = Mblock + mp * 32 + s * 16 + hi8 + r;
        zp[(size_t)b * 256 + j] = acc[s][n][r];
      }
    }
}

// --- sum 4 partials, add constW (layer0), apply OperNet ---
template <bool FIRST>
__global__ __launch_bounds__(256) void reduce_act_kernel(
    const float* __restrict__ zpart, const float* __restrict__ constW,
    const float* __restrict__ opw, float* __restrict__ xout) {
  const unsigned total = BATCH * 256;
  const unsigned stride = BATCH * 256;
  for (unsigned t = blockIdx.x * blockDim.x + threadIdx.x; t < total;
       t += gridDim.x * blockDim.x) {
    unsigned j = t & 255u;
    float z = (zpart[t] + zpart[stride + t]) +
              (zpart[2u * stride + t] + zpart[3u * stride + t]);
    if (FIRST) z += constW[j];
    float w0 = opw[j], w1 = opw[256 + j], w2 = opw[512 + j], w3 = opw[768 + j];
    xout[t] = w0 * z + w1 * sinf(z) + w2 * cosf(z) + w3 * (z * z);
  }
}

// --- out = x2 @ linP^T : [2048,256] x [256,64] ---
__global__ __launch_bounds__(256) void outproj_kernel(const float* __restrict__ x2,
                                                      const float* __restrict__ linP,
                                                      float* __restrict__ out) {
  const int wave = threadIdx.x >> 5;
  const int lane = threadIdx.x & 31;
  const int l15  = lane & 15;
  const int hi8  = (lane >> 4) * 8;
  const int koff = (lane >> 4) * 2;
  const int Mb   = blockIdx.x * 128 + wave * 16;

  const float* arow = x2 + (size_t)(Mb + l15) * 256 + koff;
  v8f acc[4] = {v8f{}, v8f{}, v8f{}, v8f{}};
  const float* bptr[4];
#pragma unroll
  for (int n = 0; n < 4; ++n)
    bptr[n] = linP + (size_t)(n * 16 + l15) * 256 + koff;

#pragma unroll 8
  for (int kk = 0; kk < 256; kk += 4) {
    v2f a = *(const v2f*)(arow + kk);
#pragma unroll
    for (int n = 0; n < 4; ++n) {
      v2f bf = *(const v2f*)(bptr[n] + kk);
      acc[n] = wmma4(a, bf, acc[n]);
    }
  }
#pragma unroll
  for (int n = 0; n < 4; ++n)
#pragma unroll
    for (int r = 0; r < 8; ++r)
      out[(size_t)(Mb + hi8 + r) * 64 + n * 16 + l15] = acc[n][r];
}

extern "C" void kernel_launch(void* const* d_in, const int* in_sizes, int n_in,
                              void* d_out, int out_size, void* d_ws, size_t ws_size,
                              hipStream_t stream) {
  (void)in_sizes; (void)n_in; (void)out_size; (void)ws_size;
  const float* x      = (const float*)d_in[0];  // [2048,128]
  const float* polyW  = (const float*)d_in[1];  // [98304,129]
  const float* constW = (const float*)d_in[2];  // [256,1]
  const float* linP   = (const float*)d_in[3];  // [64,256]
  const float* opw0   = (const float*)d_in[4];  // [4,256]
  const float* opw1   = (const float*)d_in[5];  // [4,256]
  float* out = (float*)d_out;                   // [2048,64]

  float* pmain = (float*)d_ws;                         // E*128 floats (48 MB, L2-resident)
  float* pbias = pmain + (size_t)E_TOTAL * 128;        // E floats
  float* x1    = pbias + E_TOTAL;                      // B*256 floats
  float* x2    = x1 + (size_t)BATCH * 256;             // B*256 floats
  float* zpart = x2 + (size_t)BATCH * 256;             // 4*B*256 floats (8 MB)

  repack_kernel<<<2048, 256, 0, stream>>>(polyW, pmain, pbias);

  polylayer_kernel<128><<<dim3(BATCH / 64, 4), 256, 0, stream>>>(
      x, x, pmain, pbias, zpart);
  reduce_act_kernel<true><<<512, 256, 0, stream>>>(zpart, constW, opw0, x1);

  polylayer_kernel<256><<<dim3(BATCH / 64, 4), 256, 0, stream>>>(
      x1, x, pmain + (size_t)32768 * 128, pbias + 32768, zpart);
  reduce_act_kernel<false><<<512, 256, 0, stream>>>(zpart, constW, opw1, x2);

  outproj_kernel<<<BATCH / 128, 256, 0, stream>>>(x2, linP, out);
}